// NegSimVideoHead_34729105555761
// MI455X (gfx1250) — compile-verified
//
#include <hip/hip_runtime.h>

typedef __attribute__((ext_vector_type(2))) float v2f;
typedef __attribute__((ext_vector_type(8))) float v8f;

#define NROWS 8192
#define NC    256
#define EPSN  1e-12f

// ---------------------------------------------------------------------------
// 1) Row L2-normalize: out = x / max(||x||, eps).  grid=(8192,4), block=256
// ---------------------------------------------------------------------------
__global__ __launch_bounds__(256) void normalize_kernel(
    const float* __restrict__ a0, const float* __restrict__ a1,
    const float* __restrict__ a2, const float* __restrict__ a3,
    float* __restrict__ o0, float* __restrict__ o1,
    float* __restrict__ o2, float* __restrict__ o3) {
  int row = blockIdx.x;
  int m   = blockIdx.y;
  const float* in = (m == 0) ? a0 : (m == 1) ? a1 : (m == 2) ? a2 : a3;
  float*      out = (m == 0) ? o0 : (m == 1) ? o1 : (m == 2) ? o2 : o3;
  int t = threadIdx.x;
  float x = in[row * NC + t];
  float p = x * x;
  for (int off = 16; off; off >>= 1) p += __shfl_down(p, off, 32);
  __shared__ float lds[8];
  __shared__ float stot;
  if ((t & 31) == 0) lds[t >> 5] = p;
  __syncthreads();
  if (t == 0) {
    float s = 0.f;
    for (int i = 0; i < 8; ++i) s += lds[i];
    stot = s;
  }
  __syncthreads();
  float denom = fmaxf(sqrtf(stot), EPSN);
  out[row * NC + t] = x / denom;
}

// ---------------------------------------------------------------------------
// 2) Diagonals: diag1[i] = <v1n_i, p2n_i>, diag2[i] = <v2n_i, p1n_i>
//    grid=8192, block=256
// ---------------------------------------------------------------------------
__global__ __launch_bounds__(256) void diag_kernel(
    const float* __restrict__ v1n, const float* __restrict__ v2n,
    const float* __restrict__ p1n, const float* __restrict__ p2n,
    float* __restrict__ diag1, float* __restrict__ diag2) {
  int row = blockIdx.x;
  int t = threadIdx.x;
  float a = v1n[row * NC + t] * p2n[row * NC + t];
  float b = v2n[row * NC + t] * p1n[row * NC + t];
  for (int off = 16; off; off >>= 1) {
    a += __shfl_down(a, off, 32);
    b += __shfl_down(b, off, 32);
  }
  __shared__ float la[8], lb[8];
  if ((t & 31) == 0) { la[t >> 5] = a; lb[t >> 5] = b; }
  __syncthreads();
  if (t == 0) {
    float sa = 0.f, sb = 0.f;
    for (int i = 0; i < 8; ++i) { sa += la[i]; sb += lb[i]; }
    diag1[row] = sa;
    diag2[row] = sb;
  }
}

// ---------------------------------------------------------------------------
// 3) Zero the count arrays (ws is poisoned, not re-zeroed between replays)
// ---------------------------------------------------------------------------
__global__ __launch_bounds__(256) void zero_kernel(unsigned* __restrict__ p, int n) {
  int i = blockIdx.x * 256 + threadIdx.x;
  if (i < n) p[i] = 0u;
}

// ---------------------------------------------------------------------------
// 4) Fused GEMM + rank-count using V_WMMA_F32_16X16X4_F32.
//    s = A @ B^T (A,B row-major [8192,256]); cnt[i] += #{j != i : s[i][j] > diag[i]}
//    grid=(256 rowTiles, 2 matrices), block=256 (8 waves).
//    Wave w handles cols [w*1024, w*1024+1024) for a 32-row strip, computing
//    32x32 tiles as 2x2 WMMA accumulators.
// ---------------------------------------------------------------------------
__global__ __launch_bounds__(256) void count_kernel(
    const float* __restrict__ A1, const float* __restrict__ B1,
    const float* __restrict__ A2, const float* __restrict__ B2,
    const float* __restrict__ diag1, const float* __restrict__ diag2,
    unsigned* __restrict__ cnt1, unsigned* __restrict__ cnt2) {
  const int wave = threadIdx.x >> 5;
  const int lane = threadIdx.x & 31;
  const int rowBase = blockIdx.x * 32;
  const int mat = blockIdx.y;

  const float* __restrict__ A    = mat ? A2 : A1;
  const float* __restrict__ B    = mat ? B2 : B1;
  const float* __restrict__ diag = mat ? diag2 : diag1;
  unsigned*   __restrict__ cnt   = mat ? cnt2 : cnt1;

  const int m    = lane & 15;   // M (for A) / N (for B) / col (for C)
  const int hi   = lane >> 4;   // lane-half selector
  const int koff = hi * 2;      // K offset within fragment

  // Preload diagonal values for the 16 output rows this lane touches.
  float d[2][8];
#pragma unroll
  for (int h = 0; h < 2; ++h)
#pragma unroll
    for (int r = 0; r < 8; ++r)
      d[h][r] = diag[rowBase + 16 * h + 8 * hi + r];

  unsigned c[2][8];
#pragma unroll
  for (int h = 0; h < 2; ++h)
#pragma unroll
    for (int r = 0; r < 8; ++r) c[h][r] = 0u;

  const float* aptr0 = A + (size_t)(rowBase + m) * NC + koff;
  const float* aptr1 = A + (size_t)(rowBase + 16 + m) * NC + koff;

  const int colChunk = wave * 1024;
  for (int ct = 0; ct < 32; ++ct) {
    const int colBase = colChunk + ct * 32;
    const float* bptr0 = B + (size_t)(colBase + m) * NC + koff;
    const float* bptr1 = B + (size_t)(colBase + 16 + m) * NC + koff;

    v8f acc00 = {}, acc01 = {}, acc10 = {}, acc11 = {};
#pragma unroll 4
    for (int k = 0; k < NC; k += 4) {
      v2f a0 = *(const v2f*)(aptr0 + k);
      v2f a1 = *(const v2f*)(aptr1 + k);
      v2f b0 = *(const v2f*)(bptr0 + k);
      v2f b1 = *(const v2f*)(bptr1 + k);
      acc00 = __builtin_amdgcn_wmma_f32_16x16x4_f32(false, a0, false, b0,
                                                    (short)0, acc00, false, false);
      acc01 = __builtin_amdgcn_wmma_f32_16x16x4_f32(false, a0, false, b1,
                                                    (short)0, acc01, false, false);
      acc10 = __builtin_amdgcn_wmma_f32_16x16x4_f32(false, a1, false, b0,
                                                    (short)0, acc10, false, false);
      acc11 = __builtin_amdgcn_wmma_f32_16x16x4_f32(false, a1, false, b1,
                                                    (short)0, acc11, false, false);
    }

    const int col0 = colBase + m;
    const int col1 = colBase + 16 + m;
#pragma unroll
    for (int r = 0; r < 8; ++r) {
      const int row0 = rowBase + 8 * hi + r;       // rows of acc0x
      const int row1 = row0 + 16;                  // rows of acc1x
      c[0][r] += (unsigned)((acc00[r] > d[0][r]) & (col0 != row0)) +
                 (unsigned)((acc01[r] > d[0][r]) & (col1 != row0));
      c[1][r] += (unsigned)((acc10[r] > d[1][r]) & (col0 != row1)) +
                 (unsigned)((acc11[r] > d[1][r]) & (col1 != row1));
    }
  }

  // Reduce across the 16 lanes sharing the same hi (xor masks stay in-group).
#pragma unroll
  for (int h = 0; h < 2; ++h)
#pragma unroll
    for (int r = 0; r < 8; ++r) {
      unsigned v = c[h][r];
      v += __shfl_xor(v, 1, 32);
      v += __shfl_xor(v, 2, 32);
      v += __shfl_xor(v, 4, 32);
      v += __shfl_xor(v, 8, 32);
      c[h][r] = v;
    }

  // One lane per (h, hi, r) commits the strip-chunk partial count.
#pragma unroll
  for (int h = 0; h < 2; ++h)
#pragma unroll
    for (int r = 0; r < 8; ++r)
      if (m == r)
        atomicAdd(&cnt[rowBase + 16 * h + 8 * hi + r], c[h][r]);
}

// ---------------------------------------------------------------------------
// 5) Final reduction -> 5 scalars. 1 block, 256 threads.
// ---------------------------------------------------------------------------
__global__ __launch_bounds__(256) void finalize_kernel(
    const float* __restrict__ diag1, const float* __restrict__ diag2,
    const unsigned* __restrict__ cnt1, const unsigned* __restrict__ cnt2,
    float* __restrict__ out) {
  int t = threadIdx.x;
  float sd = 0.f;
  unsigned sp = 0u, r1 = 0u, r5 = 0u, r10 = 0u;
  for (int i = t; i < NROWS; i += 256) {
    sd += diag1[i] + diag2[i];
    unsigned p1 = cnt1[i], p2 = cnt2[i];
    sp  += p1 + p2 + 1u;
    r1  += (unsigned)(p1 < 1u)  + (unsigned)(p2 < 1u);
    r5  += (unsigned)(p1 < 5u)  + (unsigned)(p2 < 5u);
    r10 += (unsigned)(p1 < 10u) + (unsigned)(p2 < 10u);
  }
  __shared__ float    lsd[256];
  __shared__ unsigned lsp[256], lr1[256], lr5[256], lr10[256];
  lsd[t] = sd; lsp[t] = sp; lr1[t] = r1; lr5[t] = r5; lr10[t] = r10;
  __syncthreads();
  for (int off = 128; off; off >>= 1) {
    if (t < off) {
      lsd[t]  += lsd[t + off];
      lsp[t]  += lsp[t + off];
      lr1[t]  += lr1[t + off];
      lr5[t]  += lr5[t + off];
      lr10[t] += lr10[t + off];
    }
    __syncthreads();
  }
  if (t == 0) {
    const float Nf = (float)NROWS;
    out[0] = -0.5f * (lsd[0] / Nf);          // neg_sim_loss
    out[1] = (float)lr1[0]  / (2.0f * Nf);   // recall1
    out[2] = (float)lr5[0]  / (2.0f * Nf);   // recall5
    out[3] = (float)lr10[0] / (2.0f * Nf);   // recall10
    out[4] = (float)lsp[0]  / (2.0f * Nf);   // mean_rk
  }
}

// ---------------------------------------------------------------------------
extern "C" void kernel_launch(void* const* d_in, const int* in_sizes, int n_in,
                              void* d_out, int out_size, void* d_ws, size_t ws_size,
                              hipStream_t stream) {
  (void)in_sizes; (void)n_in; (void)out_size; (void)ws_size;
  const float* v_feat1 = (const float*)d_in[0];
  const float* v_feat2 = (const float*)d_in[1];
  const float* p_v1    = (const float*)d_in[2];
  const float* p_v2    = (const float*)d_in[3];
  float* out = (float*)d_out;

  const size_t MATF = (size_t)NROWS * NC;  // floats per matrix
  char* ws = (char*)d_ws;
  float* v1n = (float*)(ws);
  float* v2n = v1n + MATF;
  float* p1n = v2n + MATF;
  float* p2n = p1n + MATF;
  float* diag1 = p2n + MATF;                   // 8192 floats
  float* diag2 = diag1 + NROWS;                // 8192 floats
  unsigned* cnt1 = (unsigned*)(diag2 + NROWS); // 8192 u32
  unsigned* cnt2 = cnt1 + NROWS;               // 8192 u32

  // 1) normalize all four matrices
  normalize_kernel<<<dim3(NROWS, 4), 256, 0, stream>>>(
      v_feat1, v_feat2, p_v1, p_v2, v1n, v2n, p1n, p2n);

  // 2) diagonals (also feed the loss)
  diag_kernel<<<NROWS, 256, 0, stream>>>(v1n, v2n, p1n, p2n, diag1, diag2);

  // 3) zero counts
  zero_kernel<<<(2 * NROWS + 255) / 256, 256, 0, stream>>>(cnt1, 2 * NROWS);

  // 4) fused WMMA GEMM + rank counting:  s1 = v1n @ p2n^T,  s2 = v2n @ p1n^T
  count_kernel<<<dim3(NROWS / 32, 2), 256, 0, stream>>>(
      v1n, p2n, v2n, p1n, diag1, diag2, cnt1, cnt2);

  // 5) finalize 5 scalars
  finalize_kernel<<<1, 256, 0, stream>>>(diag1, diag2, cnt1, cnt2, out);
}